// GraphSAGE_53592601920046
// MI455X (gfx1250) — compile-verified
//
#include <hip/hip_runtime.h>

#define FD 128
#define BN_EPS 1e-5f

typedef float v2f __attribute__((ext_vector_type(2)));
typedef float v8f __attribute__((ext_vector_type(8)));

__global__ void zero_f32(float* __restrict__ p, long n) {
  long i = (long)blockIdx.x * blockDim.x + threadIdx.x;
  long stride = (long)gridDim.x * blockDim.x;
  for (; i < n; i += stride) p[i] = 0.0f;
}

__global__ void count_edges(const int* __restrict__ dst, float* __restrict__ cnt, int E) {
  int i = blockIdx.x * blockDim.x + threadIdx.x;
  if (i < E) atomicAdd(&cnt[dst[i]], 1.0f);
}

__global__ void inv_counts(float* __restrict__ cnt, int n) {
  int i = blockIdx.x * blockDim.x + threadIdx.x;
  if (i < n) cnt[i] = 1.0f / fmaxf(cnt[i], 1.0f);
}

// One wave per edge: 32 lanes x float4 = 512B row gather (fully coalesced),
// then 4 global_atomic_add_f32 per lane into the L2-resident accumulator.
__global__ void scatter_rows(const float* __restrict__ x, const int* __restrict__ src,
                             const int* __restrict__ dst, float* __restrict__ agg, int E) {
  int lane = threadIdx.x & 31;
  int gw = (blockIdx.x * blockDim.x + threadIdx.x) >> 5;
  int nw = (gridDim.x * blockDim.x) >> 5;
  for (int e = gw; e < E; e += nw) {
    int s = src[e];
    int d = dst[e];
    const float4 v = *(const float4*)(x + (long)s * FD + lane * 4);
    float* o = agg + (long)d * FD + lane * 4;
    atomicAdd(o + 0, v.x);
    atomicAdd(o + 1, v.y);
    atomicAdd(o + 2, v.z);
    atomicAdd(o + 3, v.w);
  }
}

// Fused: out = BN( agg*inv_cnt @ Wl.T + x @ Wr.T + bias ), optional ReLU.
// One wave computes a 16-row tile of all 128 output features using
// V_WMMA_F32_16X16X4_F32 (fp32-exact, matches fp32 reference).
// Both A operands (agg row-tile, x row-tile) are register-cached across the
// 8 n-tiles, so the steady-state inner loop issues 1 B-load per WMMA.
template <bool RELU>
__global__ __launch_bounds__(256) void sage_gemm_bn(
    const float* __restrict__ agg, const float* __restrict__ invc,
    const float* __restrict__ xin,
    const float* __restrict__ Wl, const float* __restrict__ Wr,
    const float* __restrict__ bias, const float* __restrict__ gamma,
    const float* __restrict__ beta, const float* __restrict__ rmean,
    const float* __restrict__ rvar, float* __restrict__ out, int nrows) {
  const int lane = threadIdx.x & 31;
  const int wave = threadIdx.x >> 5;
  const int mrow = lane & 15;     // A-row / B-col within tile
  const int half = lane >> 4;     // lane group 0/1
  const int khi  = half * 2;      // K sub-offset within a 4-wide k step
  const int tile0 = (blockIdx.x * 8 + wave) * 16;
  if (tile0 >= nrows) return;     // uniform per wave

  int row  = tile0 + mrow;
  int rowc = row < nrows ? row : nrows - 1;  // clamp loads; stores guarded below
  const float* a1 = agg + (long)rowc * FD + khi;
  const float* a2 = xin + (long)rowc * FD + khi;
  const float  ic = invc[rowc];

  // Register-cache the two A fragments (invariant over n-tiles).
  // Mean-normalization of the aggregated neighbors is applied once here.
  v2f a1f[32], a2f[32];
#pragma unroll
  for (int kt = 0; kt < 32; ++kt) {
    v2f t = *(const v2f*)(a1 + kt * 4);
    a1f[kt] = t * ic;
    a2f[kt] = *(const v2f*)(a2 + kt * 4);
  }

  for (int nt = 0; nt < 8; ++nt) {
    int ncol = nt * 16 + mrow;
    const float* b1 = Wl + (long)ncol * FD + khi;  // W.T[k][n] = W[n][k]
    const float* b2 = Wr + (long)ncol * FD + khi;
    v8f c = {0.f, 0.f, 0.f, 0.f, 0.f, 0.f, 0.f, 0.f};
#pragma unroll
    for (int kt = 0; kt < 32; ++kt) {
      v2f b = *(const v2f*)(b1 + kt * 4);
      c = __builtin_amdgcn_wmma_f32_16x16x4_f32(false, a1f[kt], false, b,
                                                (short)0, c, false, false);
    }
#pragma unroll
    for (int kt = 0; kt < 32; ++kt) {
      v2f b = *(const v2f*)(b2 + kt * 4);
      c = __builtin_amdgcn_wmma_f32_16x16x4_f32(false, a2f[kt], false, b,
                                                (short)0, c, false, false);
    }
    // BN(eval) epilogue: (acc + bias - rmean) * g/sqrt(rv+eps) + beta
    float s  = gamma[ncol] * rsqrtf(rvar[ncol] + BN_EPS);
    float sh = beta[ncol] + (bias[ncol] - rmean[ncol]) * s;
#pragma unroll
    for (int r = 0; r < 8; ++r) {
      int orow = tile0 + r + half * 8;  // C/D layout: VGPR r -> M = r (+8 for hi lanes)
      float v = c[r] * s + sh;
      if (RELU) v = fmaxf(v, 0.0f);
      if (orow < nrows) out[(long)orow * FD + ncol] = v;
    }
  }
}

extern "C" void kernel_launch(void* const* d_in, const int* in_sizes, int n_in,
                              void* d_out, int out_size, void* d_ws, size_t ws_size,
                              hipStream_t stream) {
  (void)n_in; (void)out_size; (void)ws_size;
  const float* x   = (const float*)d_in[0];
  const int*   ei  = (const int*)d_in[1];
  const float* Wl0 = (const float*)d_in[2];
  const float* Wr0 = (const float*)d_in[3];
  const float* b0  = (const float*)d_in[4];
  const float* g0  = (const float*)d_in[5];
  const float* be0 = (const float*)d_in[6];
  const float* rm0 = (const float*)d_in[7];
  const float* rv0 = (const float*)d_in[8];
  const float* Wl1 = (const float*)d_in[9];
  const float* Wr1 = (const float*)d_in[10];
  const float* b1  = (const float*)d_in[11];
  const float* g1  = (const float*)d_in[12];
  const float* be1 = (const float*)d_in[13];
  const float* rm1 = (const float*)d_in[14];
  const float* rv1 = (const float*)d_in[15];

  const int N = in_sizes[0] / FD;
  const int E = in_sizes[1] / 2;
  const int* src = ei;
  const int* dst = ei + E;

  // Workspace layout (floats): cnt[N] | agg[N*FD] | h[N*FD]  (~51.4 MB)
  float* cnt = (float*)d_ws;
  float* agg = cnt + N;
  float* h   = agg + (long)N * FD;
  float* out = (float*)d_out;

  const int gb = (N + 127) / 128;  // 128 rows per workgroup (8 waves x 16 rows)

  // degree counts (same for both layers) + zero accumulator
  zero_f32<<<1024, 256, 0, stream>>>(cnt, (long)N + (long)N * FD);
  count_edges<<<(E + 255) / 256, 256, 0, stream>>>(dst, cnt, E);
  inv_counts<<<(N + 255) / 256, 256, 0, stream>>>(cnt, N);

  // layer 0: scatter-mean -> SAGE GEMM -> BN -> ReLU
  scatter_rows<<<2048, 256, 0, stream>>>(x, src, dst, agg, E);
  sage_gemm_bn<true><<<gb, 256, 0, stream>>>(agg, cnt, x, Wl0, Wr0, b0, g0, be0,
                                             rm0, rv0, h, N);

  // layer 1: re-zero accumulator, scatter-mean on h -> GEMM -> BN
  zero_f32<<<1024, 256, 0, stream>>>(agg, (long)N * FD);
  scatter_rows<<<2048, 256, 0, stream>>>(h, src, dst, agg, E);
  sage_gemm_bn<false><<<gb, 256, 0, stream>>>(agg, cnt, h, Wl1, Wr1, b1, g1, be1,
                                              rm1, rv1, out, N);
}